// PerLevelDiceLoss_75033078661123
// MI455X (gfx1250) — compile-verified
//
#include <hip/hip_runtime.h>
#include <hip/hip_bf16.h>

typedef __attribute__((ext_vector_type(2))) float v2f;
typedef __attribute__((ext_vector_type(8))) float v8f;

// Pointer types matching the async-to-LDS builtin's signature
// (diagnostic showed: __attribute__((__vector_size__(4 * sizeof(int)))) int *)
typedef int v4i_vs __attribute__((vector_size(4 * sizeof(int))));
typedef __attribute__((address_space(1))) v4i_vs* gv4i_p;   // global
typedef __attribute__((address_space(3))) v4i_vs* lv4i_p;   // LDS

#define HH 256
#define WW 256
#define KK 25
#define BB 64
#define RADIUS 14        // LENGTH-1
#define PAD 260          // LDS row pitch (floats): 1040B rows -> 8B-aligned, conflict-free b64 reads
#define NWAVE 8

#define USE_ASYNC __has_builtin(__builtin_amdgcn_global_load_async_to_lds_b128)

__global__ void dice_zero_kernel(float* out) {
  if (threadIdx.x == 0 && blockIdx.x == 0) out[0] = 0.0f;
}

__device__ __forceinline__ float sigmoidf_fast(float x) {
#if __has_builtin(__builtin_amdgcn_tanhf)
  // sigmoid(x) = 0.5 + 0.5*tanh(x/2): one TRANS op instead of exp+rcp
  return __builtin_fmaf(0.5f, __builtin_amdgcn_tanhf(0.5f * x), 0.5f);
#else
  return __builtin_amdgcn_rcpf(1.0f + __expf(-x));
#endif
}

// Stage one 16x256 f32 tile (raw logits) into a padded LDS buffer.
// Async path: GLOBAL_LOAD_ASYNC_TO_LDS_B128, 4 instructions per wave (512B each),
// no VGPR round-trip, tracked by ASYNCcnt. Fallback: coalesced float4 copy.
__device__ __forceinline__ void stage_tile(const float* __restrict__ src,
                                           float* dst, int tid) {
#if USE_ASYNC
  #pragma unroll
  for (int j = 0; j < 4; ++j) {
    const int f   = tid + j * 256;   // float4 slot index
    const int row = f >> 6;
    const int col = (f & 63) << 2;
    __builtin_amdgcn_global_load_async_to_lds_b128(
        (gv4i_p)(src + row * WW + col),
        (lv4i_p)(dst + row * PAD + col),
        0, 0);
  }
#else
  #pragma unroll
  for (int j = 0; j < 4; ++j) {
    const int f   = tid + j * 256;
    const int row = f >> 6;
    const int col = (f & 63) << 2;
    *(float4*)&dst[row * PAD + col] = *(const float4*)(src + row * WW + col);
  }
#endif
}

__device__ __forceinline__ void wait_async_le4() {
#if __has_builtin(__builtin_amdgcn_s_wait_asynccnt)
  __builtin_amdgcn_s_wait_asynccnt(4);
#elif USE_ASYNC
  asm volatile("s_wait_asynccnt 0x4" ::: "memory");
#endif
}
__device__ __forceinline__ void wait_async_le0() {
#if __has_builtin(__builtin_amdgcn_s_wait_asynccnt)
  __builtin_amdgcn_s_wait_asynccnt(0);
#elif USE_ASYNC
  asm volatile("s_wait_asynccnt 0x0" ::: "memory");
#endif
}

__launch_bounds__(256, 4)
__global__ void dice_loss_kernel(const float* __restrict__ preds,
                                 const int* __restrict__ labels,
                                 const int* __restrict__ have,
                                 float* __restrict__ out) {
  __shared__ float P[2][16 * PAD];       // double-buffered raw-logit tiles
  __shared__ float Ttab[16];             // exp(-d^2/15), d=0..14; Ttab[15]=0 (OOR sentinel)
  __shared__ float red[NWAVE][16][3];    // per-wave partials: inter, cardp, msum
  __shared__ float lossred[16];
  __shared__ int   klist[KK];
  __shared__ int   kn_sh;

  const int tid  = threadIdx.x;
  const int lane = tid & 31;
  const int wv   = tid >> 5;
  const int b    = blockIdx.y;
  const int h0   = blockIdx.x * 16;      // 16 h-rows per block

  if (tid < 16)
    Ttab[tid] = (tid <= RADIUS) ? __expf(-(float)(tid * tid) * (1.0f / 15.0f)) : 0.0f;
  if (tid == 0) {                        // block-uniform active-keypoint list
    int n = 0;
    for (int k = 0; k < KK; ++k)
      if (have[b * KK + k] != 0) klist[n++] = k;
    kn_sh = n;
  }
  __syncthreads();
  const int kn = kn_sh;

  // --- wave32 WMMA fragment indexing (f32 16x16x4) ---
  const int  arow  = lane & 15;          // A: M index
  const int  akoff = (lane >> 4) << 1;   // A: K base (0 or 2)
  const bool bcol0 = (arow == 0);        // B: we only populate column 0
  const bool cown  = (lane == 0) || (lane == 16);  // C/D col 0 owners
  const int  rbase = (lane >> 4) << 3;   // rows 0..7 vs 8..15

  v8f c1 = {0.f, 0.f, 0.f, 0.f, 0.f, 0.f, 0.f, 0.f};  // sum over all k of sum_w p
  float inter_acc[8], msum_acc[8];
  #pragma unroll
  for (int i = 0; i < 8; ++i) { inter_acc[i] = 0.f; msum_acc[i] = 0.f; }

  const int wbase = wv * 32;             // this wave owns w in [wbase, wbase+32)
  const v2f b1 = { bcol0 ? 1.0f : 0.f, bcol0 ? 1.0f : 0.f };

  if (kn > 0)
    stage_tile(preds + ((size_t)(b * KK + klist[0]) * HH + h0) * WW, P[0], tid);

  for (int i = 0; i < kn; ++i) {
    const int k = klist[i];
    float* Pbuf = P[i & 1];

    // Prefetch next tile into the other buffer (freed by the barrier ending i-1).
    if (i + 1 < kn) {
      stage_tile(preds + ((size_t)(b * KK + klist[i + 1]) * HH + h0) * WW,
                 P[(i + 1) & 1], tid);
      wait_async_le4();                  // my 4 loads for buffer i are done
    } else {
      wait_async_le0();
    }
    __syncthreads();                     // everyone's loads for buffer i are done

    const int yc = labels[(b * KK + k) * 2 + 0];  // column center
    const int xc = labels[(b * KK + k) * 2 + 1];  // row center

    v8f c2 = {0.f, 0.f, 0.f, 0.f, 0.f, 0.f, 0.f, 0.f};  // per-k: sum_w p*gy

    #pragma unroll
    for (int cc = 0; cc < 8; ++cc) {
      const int w0 = wbase + cc * 4;
      const v2f araw = *(const v2f*)&Pbuf[arow * PAD + w0 + akoff];
      v2f a;
      a.x = sigmoidf_fast(araw.x);
      a.y = sigmoidf_fast(araw.y);
      // matrix pipe reduces while VALU runs the tanh stream
      c1 = __builtin_amdgcn_wmma_f32_16x16x4_f32(false, a, false, b1, (short)0, c1, false, false);
      if (w0 + 3 >= yc - RADIUS && w0 <= yc + RADIUS) {   // wave-uniform window test
        int t0 = w0 + akoff + 0 - yc; if (t0 < 0) t0 = -t0; if (t0 > 15) t0 = 15;
        int t1 = w0 + akoff + 1 - yc; if (t1 < 0) t1 = -t1; if (t1 > 15) t1 = 15;
        const float g0 = Ttab[t0];       // unconditional LDS load, sentinel gives 0
        const float g1 = Ttab[t1];
        const v2f b2 = { bcol0 ? g0 : 0.f, bcol0 ? g1 : 0.f };
        c2 = __builtin_amdgcn_wmma_f32_16x16x4_f32(false, a, false, b2, (short)0, c2, false, false);
      }
    }

    // Per-k epilogue: scale the row-dot by gx(h); add analytic mask row-sum gx*Sgy.
    if (cown) {
      float sgy = 0.f;
      #pragma unroll
      for (int t = -RADIUS; t <= RADIUS; ++t) {
        const int w = yc + t;
        const float g = Ttab[t < 0 ? -t : t];
        sgy += (w >= 0 && w < WW) ? g : 0.f;
      }
      #pragma unroll
      for (int ii = 0; ii < 8; ++ii) {
        int dr = h0 + rbase + ii - xc; if (dr < 0) dr = -dr; if (dr > 15) dr = 15;
        const float gx = Ttab[dr];
        inter_acc[ii] = __builtin_fmaf(gx, c2[ii], inter_acc[ii]);
        msum_acc[ii]  = __builtin_fmaf(gx, sgy,    msum_acc[ii]);
      }
    }
    __syncthreads();                     // compute on buffer i done -> it may be overwritten
  }

  // Per-wave partials -> LDS
  if (cown) {
    #pragma unroll
    for (int ii = 0; ii < 8; ++ii) {
      red[wv][rbase + ii][0] = inter_acc[ii];
      red[wv][rbase + ii][1] = c1[ii];
      red[wv][rbase + ii][2] = msum_acc[ii];
    }
  }
  __syncthreads();

  // Combine waves, form dice loss per h-row
  if (tid < 16) {
    float inter = 0.f, cardp = 0.f, msum = 0.f;
    for (int wvi = 0; wvi < NWAVE; ++wvi) {
      inter += red[wvi][tid][0];
      cardp += red[wvi][tid][1];
      msum  += red[wvi][tid][2];
    }
    const float card = cardp + msum;
    const float loss = 1.0f - 2.0f * inter / fmaxf(card, 1e-7f);
    lossred[tid] = (msum > 0.f) ? loss : 0.f;   // smp masks absent classes
  }
  __syncthreads();
  if (tid == 0) {
    float s = 0.f;
    #pragma unroll
    for (int ii = 0; ii < 16; ++ii) s += lossred[ii];
    atomicAdd(out, s * (1.0f / (float)(BB * HH)));
  }
}

extern "C" void kernel_launch(void* const* d_in, const int* in_sizes, int n_in,
                              void* d_out, int out_size, void* d_ws, size_t ws_size,
                              hipStream_t stream) {
  (void)in_sizes; (void)n_in; (void)out_size; (void)d_ws; (void)ws_size;
  const float* preds  = (const float*)d_in[0];
  const int*   labels = (const int*)d_in[1];
  const int*   have   = (const int*)d_in[2];
  float* out = (float*)d_out;

  dice_zero_kernel<<<1, 32, 0, stream>>>(out);
  dim3 grid(HH / 16, BB);   // (h-tile, batch)
  dice_loss_kernel<<<grid, 256, 0, stream>>>(preds, labels, have, out);
}